// GraphSAGE_1872605741714
// MI455X (gfx1250) — compile-verified
//
#include <hip/hip_runtime.h>

typedef __attribute__((ext_vector_type(2))) float v2f;
typedef __attribute__((ext_vector_type(8))) float v8f;

#define N_SRC0 286000
#define N_DST0 11000
#define N_E0   275000
#define N_DST1 1000
#define N_E1   10000
#define IN_F   602
#define NHID   256
#define NCLS   41

// padded dims for 16x16 tiling
#define M0_PAD 11008            // 688 * 16
#define TM0    (M0_PAD / 16)    // 688
#define TN0    (NHID / 16)      // 16
#define M1_PAD 1008             // 63 * 16
#define TM1    (M1_PAD / 16)    // 63
#define TN1    3                // 48 cols >= 41
#define K0_Q   151              // ceil(602/4): k4 blocks for layer 0
#define K1_Q   64               // 256/4

// workspace layout (float offsets)
#define OFF_ACC0 0ll                       // 11008 x 602 (+64 slack for v2f over-read)
#define OFF_CNT0 6626880ll                 // 11008
#define OFF_ACC1 6637888ll                 // 1008 x 256
#define OFF_CNT1 6895936ll                 // 1008
#define ZERO_LEN 6896944ll
#define OFF_H    6896960ll                 // 11008 x 256
#define OFF_W0S  9715008ll                 // 151*2*256*2 = 154624
#define OFF_W0N  9869632ll
#define OFF_W1S  10024256ll                // 64*2*48*2 = 12288
#define OFF_W1N  10036544ll
// total = 10,048,832 floats = 40.2 MB

__global__ void zero_ws(float* ws) {
    long long i = (long long)blockIdx.x * blockDim.x + threadIdx.x;
    long long stride = (long long)gridDim.x * blockDim.x;
    for (; i < ZERO_LEN; i += stride) ws[i] = 0.0f;
}

// pack W (K x N row-major, K=602, N=256) into WMMA-B lane layout:
// out[((k4*2 + half)*256 + col)*2 + j] = (ka+j < 602) ? W[(ka+j)*256 + col] : 0
// where ka = k4*4 + 2*half
__global__ void pack_w0(const float* __restrict__ W, float* __restrict__ out) {
    int t = blockIdx.x * blockDim.x + threadIdx.x;      // over K0_Q*2*256
    if (t >= K0_Q * 2 * 256) return;
    int col  = t % 256;
    int half = (t / 256) & 1;
    int k4   = t / 512;
    int ka   = k4 * 4 + 2 * half;
    float v0 = (ka     < IN_F) ? W[(long long)ka       * NHID + col] : 0.0f;
    float v1 = (ka + 1 < IN_F) ? W[(long long)(ka + 1) * NHID + col] : 0.0f;
    out[(long long)t * 2]     = v0;
    out[(long long)t * 2 + 1] = v1;
}

// pack W1 (256 x 41) into [64][2][48][2], zero beyond col 41
__global__ void pack_w1(const float* __restrict__ W, float* __restrict__ out) {
    int t = blockIdx.x * blockDim.x + threadIdx.x;      // over K1_Q*2*48
    if (t >= K1_Q * 2 * 48) return;
    int col  = t % 48;
    int half = (t / 48) & 1;
    int k4   = t / 96;
    int ka   = k4 * 4 + 2 * half;
    float v0 = (col < NCLS) ? W[(long long)ka       * NCLS + col] : 0.0f;
    float v1 = (col < NCLS) ? W[(long long)(ka + 1) * NCLS + col] : 0.0f;
    out[(long long)t * 2]     = v0;
    out[(long long)t * 2 + 1] = v1;
}

// one wave per edge: acc0[dst] += x[src], cnt0[dst] += 1
__global__ void scatter0(const float* __restrict__ x,
                         const int* __restrict__ esrc, const int* __restrict__ edst,
                         float* __restrict__ acc, float* __restrict__ cnt) {
    int edge = blockIdx.x * (blockDim.x >> 5) + (threadIdx.x >> 5);
    int lane = threadIdx.x & 31;
    if (edge >= N_E0) return;
    int s = esrc[edge], d = edst[edge];
    const float* xr = x + (long long)s * IN_F;
    float* ar = acc + (long long)d * IN_F;
    for (int f = lane; f < IN_F; f += 32) atomicAdd(&ar[f], xr[f]);
    if (lane == 0) atomicAdd(&cnt[d], 1.0f);
}

__global__ void finalize0(float* __restrict__ acc, const float* __restrict__ cnt) {
    long long i = (long long)blockIdx.x * blockDim.x + threadIdx.x;
    if (i >= (long long)N_DST0 * IN_F) return;
    float c = cnt[i / IN_F];
    acc[i] = acc[i] / fmaxf(c, 1.0f);
}

// h = relu(x[:N_DST0] @ Wself0 + agg @ Wneigh0 + b0)   -> h padded M0_PAD x NHID
// A loads: unconditional v2f (rows < 11008 always valid memory; padded rows
// discarded at store). K remainder handled by one extra WMMA with cndmask-zeroed A.
__global__ void gemm1(const float* __restrict__ x, const float* __restrict__ agg,
                      const float* __restrict__ Bs, const float* __restrict__ Bn,
                      const float* __restrict__ b0, float* __restrict__ h) {
    int wave = blockIdx.x * (blockDim.x >> 5) + (threadIdx.x >> 5);
    if (wave >= TM0 * TN0) return;                 // wave-uniform exit
    int tm = wave / TN0, tn = wave % TN0;
    int lane = threadIdx.x & 31;
    int half = lane >> 4;                          // 0: K0/K1, 1: K2/K3
    int l15  = lane & 15;
    int rowA = tm * 16 + l15;
    int colB = tn * 16 + l15;
    int boff = half * 512 + colB * 2;              // offset within a k4 block (1024 floats)

    const float* xr = x   + (long long)rowA * IN_F;
    const float* gr = agg + (long long)rowA * IN_F;

    v8f c0 = {}, c1 = {};
    // ---- self part ----
    for (int k4 = 0; k4 < 150; k4 += 2) {
        v2f a0 = *(const v2f*)(xr + k4 * 4 + 2 * half);
        v2f a1 = *(const v2f*)(xr + k4 * 4 + 4 + 2 * half);
        v2f w0 = *(const v2f*)(Bs + k4 * 1024 + boff);
        v2f w1 = *(const v2f*)(Bs + (k4 + 1) * 1024 + boff);
        c0 = __builtin_amdgcn_wmma_f32_16x16x4_f32(false, a0, false, w0, (short)0, c0, false, false);
        c1 = __builtin_amdgcn_wmma_f32_16x16x4_f32(false, a1, false, w1, (short)0, c1, false, false);
    }
    {   // K remainder: ka = 600..601 valid only for half==0
        v2f al = *(const v2f*)(xr + 600 + 2 * half);   // over-read ok (valid memory)
        v2f ar; ar.x = half ? 0.0f : al.x; ar.y = half ? 0.0f : al.y;
        v2f wr = *(const v2f*)(Bs + 150 * 1024 + boff);  // zero-padded in pack
        c0 = __builtin_amdgcn_wmma_f32_16x16x4_f32(false, ar, false, wr, (short)0, c0, false, false);
    }
    // ---- neigh part ----
    for (int k4 = 0; k4 < 150; k4 += 2) {
        v2f a0 = *(const v2f*)(gr + k4 * 4 + 2 * half);
        v2f a1 = *(const v2f*)(gr + k4 * 4 + 4 + 2 * half);
        v2f w0 = *(const v2f*)(Bn + k4 * 1024 + boff);
        v2f w1 = *(const v2f*)(Bn + (k4 + 1) * 1024 + boff);
        c0 = __builtin_amdgcn_wmma_f32_16x16x4_f32(false, a0, false, w0, (short)0, c0, false, false);
        c1 = __builtin_amdgcn_wmma_f32_16x16x4_f32(false, a1, false, w1, (short)0, c1, false, false);
    }
    {
        v2f al = *(const v2f*)(gr + 600 + 2 * half);   // acc0 has +slack, reads are legal
        v2f ar; ar.x = half ? 0.0f : al.x; ar.y = half ? 0.0f : al.y;
        v2f wr = *(const v2f*)(Bn + 150 * 1024 + boff);
        c0 = __builtin_amdgcn_wmma_f32_16x16x4_f32(false, ar, false, wr, (short)0, c0, false, false);
    }
    c0 = c0 + c1;
    float bias = b0[colB];
    #pragma unroll
    for (int v = 0; v < 8; ++v) {
        int m = tm * 16 + v + 8 * half;
        if (m < N_DST0)
            h[(long long)m * NHID + colB] = fmaxf(c0[v] + bias, 0.0f);
    }
}

// one wave per edge: acc1[dst] += h[src], cnt1[dst] += 1
__global__ void scatter1(const float* __restrict__ h,
                         const int* __restrict__ esrc, const int* __restrict__ edst,
                         float* __restrict__ acc, float* __restrict__ cnt) {
    int edge = blockIdx.x * (blockDim.x >> 5) + (threadIdx.x >> 5);
    int lane = threadIdx.x & 31;
    if (edge >= N_E1) return;
    int s = esrc[edge], d = edst[edge];
    const float* hr = h + (long long)s * NHID;
    float* ar = acc + (long long)d * NHID;
    #pragma unroll
    for (int f = lane; f < NHID; f += 32) atomicAdd(&ar[f], hr[f]);
    if (lane == 0) atomicAdd(&cnt[d], 1.0f);
}

__global__ void finalize1(float* __restrict__ acc, const float* __restrict__ cnt) {
    int i = blockIdx.x * blockDim.x + threadIdx.x;
    if (i >= N_DST1 * NHID) return;
    float c = cnt[i / NHID];
    acc[i] = acc[i] / fmaxf(c, 1.0f);
}

// out = h[:N_DST1] @ Wself1 + agg1 @ Wneigh1 + b1     (1000 x 41)
__global__ void gemm2(const float* __restrict__ h, const float* __restrict__ agg1,
                      const float* __restrict__ Bs, const float* __restrict__ Bn,
                      const float* __restrict__ b1, float* __restrict__ out) {
    int wave = blockIdx.x * (blockDim.x >> 5) + (threadIdx.x >> 5);
    if (wave >= TM1 * TN1) return;                 // wave-uniform exit
    int tm = wave / TN1, tn = wave % TN1;
    int lane = threadIdx.x & 31;
    int half = lane >> 4;
    int l15  = lane & 15;
    int rowA = tm * 16 + l15;                      // < 1008: valid rows of h / acc1
    int colB = tn * 16 + l15;                      // < 48
    bool colOK = colB < NCLS;
    int boff = half * 96 + colB * 2;               // within a k4 block (192 floats)

    const float* hr = h    + (long long)rowA * NHID;
    const float* gr = agg1 + (long long)rowA * NHID;

    v8f c0 = {}, c1 = {};
    for (int k4 = 0; k4 < K1_Q; k4 += 2) {
        v2f a0 = *(const v2f*)(hr + k4 * 4 + 2 * half);
        v2f a1 = *(const v2f*)(hr + k4 * 4 + 4 + 2 * half);
        v2f w0 = *(const v2f*)(Bs + k4 * 192 + boff);
        v2f w1 = *(const v2f*)(Bs + (k4 + 1) * 192 + boff);
        c0 = __builtin_amdgcn_wmma_f32_16x16x4_f32(false, a0, false, w0, (short)0, c0, false, false);
        c1 = __builtin_amdgcn_wmma_f32_16x16x4_f32(false, a1, false, w1, (short)0, c1, false, false);
    }
    for (int k4 = 0; k4 < K1_Q; k4 += 2) {
        v2f a0 = *(const v2f*)(gr + k4 * 4 + 2 * half);
        v2f a1 = *(const v2f*)(gr + k4 * 4 + 4 + 2 * half);
        v2f w0 = *(const v2f*)(Bn + k4 * 192 + boff);
        v2f w1 = *(const v2f*)(Bn + (k4 + 1) * 192 + boff);
        c0 = __builtin_amdgcn_wmma_f32_16x16x4_f32(false, a0, false, w0, (short)0, c0, false, false);
        c1 = __builtin_amdgcn_wmma_f32_16x16x4_f32(false, a1, false, w1, (short)0, c1, false, false);
    }
    c0 = c0 + c1;
    float bias = colOK ? b1[colB] : 0.0f;
    #pragma unroll
    for (int v = 0; v < 8; ++v) {
        int m = tm * 16 + v + 8 * half;
        if (m < N_DST1 && colB < NCLS)
            out[(long long)m * NCLS + colB] = c0[v] + bias;
    }
}

extern "C" void kernel_launch(void* const* d_in, const int* in_sizes, int n_in,
                              void* d_out, int out_size, void* d_ws, size_t ws_size,
                              hipStream_t stream) {
    const float* x      = (const float*)d_in[0];
    const float* Wself0 = (const float*)d_in[1];
    const float* Wneigh0= (const float*)d_in[2];
    const float* b0     = (const float*)d_in[3];
    const float* Wself1 = (const float*)d_in[4];
    const float* Wneigh1= (const float*)d_in[5];
    const float* b1     = (const float*)d_in[6];
    const int* e0_src   = (const int*)d_in[7];
    const int* e0_dst   = (const int*)d_in[8];
    const int* e1_src   = (const int*)d_in[9];
    const int* e1_dst   = (const int*)d_in[10];

    float* ws   = (float*)d_ws;
    float* acc0 = ws + OFF_ACC0;
    float* cnt0 = ws + OFF_CNT0;
    float* acc1 = ws + OFF_ACC1;
    float* cnt1 = ws + OFF_CNT1;
    float* h    = ws + OFF_H;
    float* w0s  = ws + OFF_W0S;
    float* w0n  = ws + OFF_W0N;
    float* w1s  = ws + OFF_W1S;
    float* w1n  = ws + OFF_W1N;
    float* out  = (float*)d_out;

    // 1) zero accumulators + counters
    zero_ws<<<2048, 256, 0, stream>>>(ws);

    // 2) pack weights into WMMA-B lane layout (coalesced, zero-padded)
    pack_w0<<<(K0_Q * 2 * 256 + 255) / 256, 256, 0, stream>>>(Wself0,  w0s);
    pack_w0<<<(K0_Q * 2 * 256 + 255) / 256, 256, 0, stream>>>(Wneigh0, w0n);
    pack_w1<<<(K1_Q * 2 * 48 + 255) / 256, 256, 0, stream>>>(Wself1,  w1s);
    pack_w1<<<(K1_Q * 2 * 48 + 255) / 256, 256, 0, stream>>>(Wneigh1, w1n);

    // 3) layer-0 scatter: 8 edges per 256-thread block (1 wave/edge)
    scatter0<<<(N_E0 + 7) / 8, 256, 0, stream>>>(x, e0_src, e0_dst, acc0, cnt0);

    // 4) divide by counts
    {
        long long tot = (long long)N_DST0 * IN_F;
        finalize0<<<(unsigned)((tot + 255) / 256), 256, 0, stream>>>(acc0, cnt0);
    }

    // 5) GEMM1 + bias + relu (WMMA f32 16x16x4): 688*16 tiles, 8 waves/block
    gemm1<<<(TM0 * TN0 + 7) / 8, 256, 0, stream>>>(x, acc0, w0s, w0n, b0, h);

    // 6) layer-1 scatter
    scatter1<<<(N_E1 + 7) / 8, 256, 0, stream>>>(h, e1_src, e1_dst, acc1, cnt1);

    // 7) divide by counts
    finalize1<<<(N_DST1 * NHID + 255) / 256, 256, 0, stream>>>(acc1, cnt1);

    // 8) GEMM2 + bias (WMMA f32 16x16x4): 63*3 tiles
    gemm2<<<(TM1 * TN1 + 7) / 8, 256, 0, stream>>>(h, acc1, w1s, w1n, b1, out);
}